// Linear_base_1_82600811037329
// MI455X (gfx1250) — compile-verified
//
#include <hip/hip_runtime.h>
#include <stdint.h>

// Problem constants
#define S_LEN     262144
#define NDIM      128
#define TILE_ROWS 128
#define AST       132                    // padded LDS row stride (floats); 132%64=4 -> conflict-free
#define NBLK      (S_LEN / TILE_ROWS)    // 2048

typedef __attribute__((ext_vector_type(2))) float v2f;
typedef __attribute__((ext_vector_type(8))) float v8f;

// Low 32 bits of a generic pointer to __shared__ data are the LDS byte offset
__device__ __forceinline__ unsigned lds_off_of(const void* p) {
  return (unsigned)(uintptr_t)p;
}

// CDNA5 async global->LDS copy (tracked by ASYNCcnt). saddr form:
// global address = SGPR64 base + VGPR32 offset; vdst VGPR holds LDS byte address.
__device__ __forceinline__ void async_ld_b128(unsigned lds_off, unsigned goff, const void* sbase) {
  asm volatile("global_load_async_to_lds_b128 %0, %1, %2"
               :: "v"(lds_off), "v"(goff), "s"(sbase) : "memory");
}
__device__ __forceinline__ void wait_async0() {
  asm volatile("s_wait_asynccnt 0" ::: "memory");
}

__device__ __forceinline__ float softplus_f(float x) {
  // stable softplus: max(x,0) + log1p(exp(-|x|))
  return fmaxf(x, 0.f) + log1pf(expf(-fabsf(x)));
}

// ---------------------------------------------------------------------------
// Kernel 1: Wcm[n*128+k] = (W_K^T @ W_Q)[k][n] = sum_c W_K[c,k] * W_Q[c,n]
// Column-major storage so B fragments in kernel 2 are contiguous-in-K.
// 64 blocks x 256 threads; one output element per thread, coalesced on k.
// ---------------------------------------------------------------------------
__global__ __launch_bounds__(256) void nk_prep(const float* __restrict__ WQ,
                                               const float* __restrict__ WK,
                                               float* __restrict__ Wcm) {
  const int idx = blockIdx.x * 256 + threadIdx.x;  // 0..16383
  const int n = idx >> 7;
  const int k = idx & 127;
  float s = 0.f;
  for (int c = 0; c < NDIM; ++c)
    s = fmaf(WK[c * NDIM + k], WQ[c * NDIM + n], s);
  Wcm[n * NDIM + k] = s;
}

// ---------------------------------------------------------------------------
// Kernel 2: per 128-row tile: V = A @ W via V_WMMA_F32_16X16X4_F32,
// aw = softplus(V), out = aw * rowsum(A), accumulate (out - A[s+1])^2.
// ---------------------------------------------------------------------------
__global__ __launch_bounds__(256) void nk_main(const float* __restrict__ In,
                                               const float* __restrict__ Wcm,
                                               float* __restrict__ partial) {
  __shared__ __align__(16) float lds_W[NDIM * AST];              // [n][k] padded
  __shared__ __align__(16) float lds_A[(TILE_ROWS + 1) * AST];   // rows r0..r0+128
  __shared__ float lds_rs[TILE_ROWS + 4];
  __shared__ float lds_red[8];

  const int tid = threadIdx.x;
  const int r0  = blockIdx.x * TILE_ROWS;

  // Async-stage W (16B granules, 4096 of them) into padded LDS
  for (int i = tid; i < NDIM * (NDIM / 4); i += 256) {
    const int n = i >> 5, c4 = i & 31;
    async_ld_b128(lds_off_of(&lds_W[n * AST + c4 * 4]), (unsigned)(i * 16), Wcm);
  }
  // Async-stage 129 Input rows (row 128 of the last tile is OOB -> skipped;
  // its garbage is only read on loss terms that are masked out below).
  const float* Abase = In + (size_t)r0 * NDIM;
  for (int i = tid; i < (TILE_ROWS + 1) * (NDIM / 4); i += 256) {
    const int row = i >> 5, c4 = i & 31;
    if (r0 + row < S_LEN)
      async_ld_b128(lds_off_of(&lds_A[row * AST + c4 * 4]),
                    (unsigned)((row * NDIM + c4 * 4) * 4), Abase);
  }
  wait_async0();
  __syncthreads();

  // Row sums of the tile (threads 0..128)
  if (tid <= TILE_ROWS) {
    float s = 0.f;
    const float* rp = &lds_A[tid * AST];
    for (int k = 0; k < NDIM; ++k) s += rp[k];
    lds_rs[tid] = s;
  }
  __syncthreads();

  const int wave = tid >> 5;
  const int lane = tid & 31;
  const int half = lane >> 4;   // 0: K even pair base 0/1, 1: K 2/3
  const int lrow = lane & 15;

  // A fragments per ISA 16x4 f32 layout: lanes 0-15 hold M=0..15 {K0,K1},
  // lanes 16-31 hold M=0..15 {K2,K3}. 32 K-steps cover K=0..127.
  v2f afrag[32];
  {
    const float* ap = &lds_A[(wave * 16 + lrow) * AST + 2 * half];
#pragma unroll
    for (int ks = 0; ks < 32; ++ks) {
      afrag[ks][0] = ap[4 * ks];
      afrag[ks][1] = ap[4 * ks + 1];
    }
  }

  float loss_acc = 0.f;
  const float* wp0 = &lds_W[lrow * AST + 2 * half];

  for (int nt = 0; nt < 8; ++nt) {
    v8f acc = {0.f, 0.f, 0.f, 0.f, 0.f, 0.f, 0.f, 0.f};
    const float* wp = wp0 + nt * 16 * AST;
#pragma unroll
    for (int ks = 0; ks < 32; ++ks) {
      v2f b;
      b[0] = wp[4 * ks];
      b[1] = wp[4 * ks + 1];
      acc = __builtin_amdgcn_wmma_f32_16x16x4_f32(
          /*neg_a=*/false, afrag[ks], /*neg_b=*/false, b,
          /*c_mod=*/(short)0, acc, /*reuse_a=*/false, /*reuse_b=*/false);
    }
    // Epilogue from C/D layout: VGPR v -> M = v + 8*half, N = lane%16
    const int ncol = nt * 16 + lrow;
#pragma unroll
    for (int v = 0; v < 8; ++v) {
      const int m = wave * 16 + 8 * half + v;       // local row 0..127
      const float x  = acc[v];
      const float o  = softplus_f(x) * lds_rs[m];
      const float t  = lds_A[(m + 1) * AST + ncol]; // Input[r0+m+1][ncol]
      const float d  = o - t;
      if (r0 + m < S_LEN - 1) loss_acc += d * d;    // mask last global row
    }
  }

  // Deterministic in-wave + in-block reduction, one partial per block
#pragma unroll
  for (int off = 16; off > 0; off >>= 1)
    loss_acc += __shfl_down(loss_acc, off, 32);
  if (lane == 0) lds_red[wave] = loss_acc;
  __syncthreads();
  if (tid == 0) {
    float s = 0.f;
#pragma unroll
    for (int w = 0; w < 8; ++w) s += lds_red[w];
    partial[blockIdx.x] = s;
  }
}

// ---------------------------------------------------------------------------
// Kernel 3: deterministic final reduction: loss mean + l1 = sum |sigmoid(W)|
// ---------------------------------------------------------------------------
__global__ __launch_bounds__(256) void nk_reduce(const float* __restrict__ partial,
                                                 const float* __restrict__ Wcm,
                                                 float* __restrict__ out) {
  __shared__ float red[256];
  const int tid = threadIdx.x;

  float s = 0.f;
  for (int i = tid; i < NBLK; i += 256) s += partial[i];
  float l1 = 0.f;
  for (int i = tid; i < NDIM * NDIM; i += 256) {
    const float w = Wcm[i];
    l1 += fabsf(1.f / (1.f + expf(-w)));
  }

  red[tid] = s;
  __syncthreads();
  for (int st = 128; st > 0; st >>= 1) {
    if (tid < st) red[tid] += red[tid + st];
    __syncthreads();
  }
  const float loss = red[0];
  __syncthreads();

  red[tid] = l1;
  __syncthreads();
  for (int st = 128; st > 0; st >>= 1) {
    if (tid < st) red[tid] += red[tid + st];
    __syncthreads();
  }
  if (tid == 0) {
    out[0] = loss / ((float)(S_LEN - 1) * (float)NDIM);
    out[1] = red[0];
  }
}

extern "C" void kernel_launch(void* const* d_in, const int* in_sizes, int n_in,
                              void* d_out, int out_size, void* d_ws, size_t ws_size,
                              hipStream_t stream) {
  (void)in_sizes; (void)n_in; (void)out_size; (void)ws_size;
  const float* In = (const float*)d_in[0];   // Input  (S, N)
  const float* WQ = (const float*)d_in[1];   // W_Q    (N, N)
  const float* WK = (const float*)d_in[2];   // W_K    (N, N)
  float* out = (float*)d_out;                // [loss, l1]
  float* Wcm = (float*)d_ws;                 // 128*128 floats (column-major W)
  float* partial = Wcm + NDIM * NDIM;        // 2048 floats

  nk_prep  <<<NDIM * NDIM / 256, 256, 0, stream>>>(WQ, WK, Wcm);
  nk_main  <<<NBLK, 256, 0, stream>>>(In, Wcm, partial);
  nk_reduce<<<1, 256, 0, stream>>>(partial, Wcm, out);
}